// SelfAttention_89524298318171
// MI455X (gfx1250) — compile-verified
//
#include <hip/hip_runtime.h>
#include <stdint.h>

typedef __attribute__((ext_vector_type(2))) float v2f;
typedef __attribute__((ext_vector_type(8))) float v8f;
typedef __attribute__((ext_vector_type(4))) unsigned int u32x4;
typedef __attribute__((ext_vector_type(8))) int i32x8;
typedef __attribute__((ext_vector_type(4))) int i32x4;

#define KDIM 256
#define TSEQ 512
#define BATCH 4
#define NCHUNK (TSEQ / 16)
#define LSTRIDE 260   // 256 + 4 pad; banks (4k+d)%64 distinct for k=0..15
#define PSTRIDE 20    // banks (20m+c)%64 distinct for m=0..15

#if __has_builtin(__builtin_amdgcn_tensor_load_to_lds)
#define USE_TDM 1
#else
#define USE_TDM 0
#endif

__device__ __forceinline__ float fast_tanh(float x) {
#if __has_builtin(__builtin_amdgcn_tanhf)
  return __builtin_amdgcn_tanhf(x);
#else
  float r;
  asm("v_tanh_f32 %0, %1" : "=v"(r) : "v"(x));
  return r;
#endif
}

__device__ __forceinline__ float fast_exp2(float x) {
  return __builtin_amdgcn_exp2f(x);
}

#if USE_TDM
// DMA one 16x256 f32 tile (row-major, row stride 256 floats in memory) into
// LDS at lds_byte_addr, inserting 4 pad DWORDs after every 256 DWORDs so the
// LDS row stride becomes LSTRIDE=260 floats (bank-conflict-free).
__device__ __forceinline__ void tdm_load_tile(const float* gsrc,
                                              uint32_t lds_byte_addr) {
  uint64_t ga = (uint64_t)(uintptr_t)gsrc;
  u32x4 g0;
  g0[0] = 1u;                                               // count=1 (valid)
  g0[1] = lds_byte_addr;                                    // lds_addr
  g0[2] = (uint32_t)ga;                                     // global_addr lo
  g0[3] = (uint32_t)((ga >> 32) & 0x1FFFFFFu) | (2u << 30); // addr[56:32]|type=2
  i32x8 g1;
  // data_size=2 (4B) | pad_enable | pad_interval=7 (256 dw) | pad_amount=3 (4 dw)
  g1[0] = (int)((2u << 16) | (1u << 20) | (7u << 22) | (3u << 25));
  g1[1] = (int)(256u << 16);   // [63:48] tensor_dim0[15:0] = 256
  g1[2] = (int)(16u << 16);    // [95:80] tensor_dim1[15:0] = 16
  g1[3] = (int)(256u << 16);   // [127:112] tile_dim0 = 256
  g1[4] = 16;                  // [143:128] tile_dim1 = 16; tile_dim2 = 0
  g1[5] = 256;                 // tensor_dim0_stride[31:0] = 256 elements
  g1[6] = 0;
  g1[7] = 0;
  i32x4 z4 = {0, 0, 0, 0};     // 2D tile: groups 2/3 unused
  i32x8 z8 = {0, 0, 0, 0, 0, 0, 0, 0};
  __builtin_amdgcn_tensor_load_to_lds(g0, g1, z4, z4, z8, 0);
}
#endif

// ---------------------------------------------------------------------------
// Kernel 1: key = h_e @ W_a^T, query = h_e @ V_a^T using V_WMMA_F32_16X16X4_F32
// One 16x16 output tile per wave, K-loop of 64 WMMAs (K=4 each).
// ---------------------------------------------------------------------------
__global__ __launch_bounds__(256) void proj_kernel(
    const float* __restrict__ X,    // [2048, 256] flattened h_e
    const float* __restrict__ Wk,   // W_a [256,256]
    const float* __restrict__ Wq,   // V_a [256,256]
    float* __restrict__ Yk,         // key_vec [2048,256]
    float* __restrict__ Yq)         // query_vec [2048,256]
{
  const int lane = threadIdx.x & 31;
  const int wave = threadIdx.x >> 5;
  const int tile = blockIdx.x * 8 + wave;   // 4096 wave-tiles total
  const int which = tile >> 11;             // 0: key (W_a), 1: query (V_a)
  const int rem = tile & 2047;
  const int ntile = rem >> 4;               // 0..127
  const int etile = rem & 15;               // 0..15
  const float* W = which ? Wq : Wk;
  float* Y = which ? Yq : Yk;
  const int n0 = ntile * 16, e0 = etile * 16;
  const int hi = lane >> 4;                 // half-wave: K offset 0 or 2
  const int m  = lane & 15;                 // row (A) / col (B,C)

  v8f acc = {};
  const float* arow = X + (n0 + m) * KDIM + 2 * hi;  // A[M][K]=X[n0+M][k+K]
  const float* brow = W + (e0 + m) * KDIM + 2 * hi;  // B[K][N]=W[e0+N][k+K]

#pragma unroll 8
  for (int k = 0; k < KDIM; k += 4) {
    v2f a  = *(const v2f*)(arow + k);
    v2f bb = *(const v2f*)(brow + k);
    acc = __builtin_amdgcn_wmma_f32_16x16x4_f32(
        false, a, false, bb, (short)0, acc, false, false);
  }

  float* orow = Y + n0 * KDIM + e0;
#pragma unroll
  for (int r = 0; r < 8; ++r)
    orow[(r + hi * 8) * KDIM + m] = acc[r];
}

// ---------------------------------------------------------------------------
// Kernel 2: fused additive-attention. One block per (batch, 16-query tile).
// Streaming softmax over 32 key chunks. TDM double-buffers the key/h_e chunk
// loads (global->LDS DMA, TENSORcnt) so the DMA of chunk kc+1 overlaps the
// tanh/softmax/WMMA work on chunk kc.
// ---------------------------------------------------------------------------
__global__ __launch_bounds__(256) void attn_kernel(
    const float* __restrict__ h_e,   // [B,T,KD]
    const float* __restrict__ keyv,  // [B,T,KD]
    const float* __restrict__ qryv,  // [B,T,KD]
    const float* __restrict__ w_a,   // [KD]
    float* __restrict__ out)         // [B,T,KD]
{
  __shared__ float qs[16 * LSTRIDE];      // query tile
  __shared__ float ks[2][16 * LSTRIDE];   // key chunk (double buffer)
  __shared__ float hs[2][16 * LSTRIDE];   // h_e chunk (double buffer)
  __shared__ float ps[16 * PSTRIDE];      // P tile (softmax numerators)
  __shared__ float wsh[KDIM];
  __shared__ float m_run[16], l_run[16], alphas[16];

  const int tid = threadIdx.x;
  const int bb = blockIdx.x >> 5;      // batch
  const int qt = blockIdx.x & 31;      // query tile
  const int q0 = qt * 16;
  const int lane = tid & 31;
  const int wave = tid >> 5;
  const int hi = lane >> 4;
  const int m = lane & 15;

  const float* qb = qryv + (bb * TSEQ + q0) * KDIM;
  const float* kb = keyv + bb * TSEQ * KDIM;
  const float* hb = h_e + bb * TSEQ * KDIM;

  // stage query tile, w_a, init running stats
  for (int i = tid; i < 16 * 64; i += 256) {
    int r = i >> 6, c = (i & 63) << 2;
    *(float4*)(&qs[r * LSTRIDE + c]) = *(const float4*)(qb + r * KDIM + c);
  }
  wsh[tid] = w_a[tid];
  if (tid < 16) { m_run[tid] = -1e30f; l_run[tid] = 0.f; }

#if USE_TDM
  const uint32_t ks_addr[2] = {(uint32_t)(uintptr_t)&ks[0][0],
                               (uint32_t)(uintptr_t)&ks[1][0]};
  const uint32_t hs_addr[2] = {(uint32_t)(uintptr_t)&hs[0][0],
                               (uint32_t)(uintptr_t)&hs[1][0]};
  if (wave == 0) {   // prologue: DMA chunk 0 into buffer 0
    tdm_load_tile(kb, ks_addr[0]);
    tdm_load_tile(hb, hs_addr[0]);
  }
#endif

  const int qrow_i = tid >> 4;   // query row handled by this thread (logits)
  const int krow_i = tid & 15;   // key col handled by this thread

  v8f acc0 = {};                 // d-slice [d0, d0+16)
  v8f acc1 = {};                 // d-slice [d0+16, d0+32)
  const int d0 = wave * 32;

  for (int kc = 0; kc < NCHUNK; ++kc) {
    const int buf = kc & 1;
    __syncthreads();   // prev iteration done with both buffers / ps / alphas
#if USE_TDM
    if (wave == 0) {
      if (kc + 1 < NCHUNK) {
        const int nb = (kc + 1) & 1;
        tdm_load_tile(kb + (size_t)(kc + 1) * 16 * KDIM, ks_addr[nb]);
        tdm_load_tile(hb + (size_t)(kc + 1) * 16 * KDIM, hs_addr[nb]);
        __builtin_amdgcn_s_wait_tensorcnt((short)2);  // chunk kc's 2 ops done
      } else {
        __builtin_amdgcn_s_wait_tensorcnt((short)0);
      }
    }
#else
    {
      const int k0 = kc * 16;
      for (int i = tid; i < 16 * 64; i += 256) {
        int r = i >> 6, c = (i & 63) << 2;
        *(float4*)(&ks[buf][r * LSTRIDE + c]) =
            *(const float4*)(kb + (k0 + r) * KDIM + c);
        *(float4*)(&hs[buf][r * LSTRIDE + c]) =
            *(const float4*)(hb + (k0 + r) * KDIM + c);
      }
    }
#endif
    __syncthreads();   // chunk kc visible to all waves

    // ---- logits: s(q,k) = sum_d w[d]*tanh(query[q,d]+key[k,d]) ----
    const float* qrow = &qs[qrow_i * LSTRIDE];       // broadcast across lanes
    const float* krow = &ks[buf][krow_i * LSTRIDE];  // conflict-free (pad 260)
    float s = 0.f;
#pragma unroll 4
    for (int d = 0; d < KDIM; d += 4) {
      float4 qv = *(const float4*)(qrow + d);
      float4 kv = *(const float4*)(krow + d);
      float4 wv = *(const float4*)(&wsh[d]);
      s += wv.x * fast_tanh(qv.x + kv.x);
      s += wv.y * fast_tanh(qv.y + kv.y);
      s += wv.z * fast_tanh(qv.z + kv.z);
      s += wv.w * fast_tanh(qv.w + kv.w);
    }

    // ---- online softmax over this 16-key chunk (16-lane groups) ----
    float rmax = s;
    rmax = fmaxf(rmax, __shfl_xor(rmax, 1));
    rmax = fmaxf(rmax, __shfl_xor(rmax, 2));
    rmax = fmaxf(rmax, __shfl_xor(rmax, 4));
    rmax = fmaxf(rmax, __shfl_xor(rmax, 8));
    const float L2E = 1.44269504088896f;
    float m_old = m_run[qrow_i];            // read before lockstep write below
    float m_new = fmaxf(m_old, rmax);
    float p = fast_exp2((s - m_new) * L2E);
    float rsum = p;
    rsum += __shfl_xor(rsum, 1);
    rsum += __shfl_xor(rsum, 2);
    rsum += __shfl_xor(rsum, 4);
    rsum += __shfl_xor(rsum, 8);
    float alpha = fast_exp2((m_old - m_new) * L2E);
    if (krow_i == 0) {
      m_run[qrow_i] = m_new;
      l_run[qrow_i] = l_run[qrow_i] * alpha + rsum;
      alphas[qrow_i] = alpha;
    }
    ps[qrow_i * PSTRIDE + krow_i] = p;
    __syncthreads();

    // ---- rescale accumulators, then acc += P[16x16] @ H[16x32-slice] ----
#pragma unroll
    for (int r = 0; r < 8; ++r) {
      float al = alphas[r + hi * 8];
      acc0[r] *= al;
      acc1[r] *= al;
    }
    const float* hrow = &hs[buf][0];
#pragma unroll
    for (int c = 0; c < 16; c += 4) {
      // A[M][K] = ps[M][c+K]: lane holds (K0,K1)/(K2,K3) contiguous -> float2
      v2f a = *(const v2f*)(&ps[m * PSTRIDE + c + 2 * hi]);
      v2f b0, b1;
      b0[0] = hrow[(c + 2 * hi)     * LSTRIDE + d0 + m];
      b0[1] = hrow[(c + 2 * hi + 1) * LSTRIDE + d0 + m];
      b1[0] = hrow[(c + 2 * hi)     * LSTRIDE + d0 + 16 + m];
      b1[1] = hrow[(c + 2 * hi + 1) * LSTRIDE + d0 + 16 + m];
      acc0 = __builtin_amdgcn_wmma_f32_16x16x4_f32(
          false, a, false, b0, (short)0, acc0, false, false);
      acc1 = __builtin_amdgcn_wmma_f32_16x16x4_f32(
          false, a, false, b1, (short)0, acc1, false, false);
    }
  }

  // ---- epilogue: divide by softmax denominator, store ----
  float* ob = out + (bb * TSEQ + q0) * KDIM;
#pragma unroll
  for (int r = 0; r < 8; ++r) {
    int row = r + hi * 8;
    float inv = 1.0f / l_run[row];
    ob[row * KDIM + d0 + m]      = acc0[r] * inv;
    ob[row * KDIM + d0 + 16 + m] = acc1[r] * inv;
  }
}

// ---------------------------------------------------------------------------
extern "C" void kernel_launch(void* const* d_in, const int* in_sizes, int n_in,
                              void* d_out, int out_size, void* d_ws, size_t ws_size,
                              hipStream_t stream) {
  (void)in_sizes; (void)n_in; (void)out_size; (void)ws_size;
  const float* h_e = (const float*)d_in[0];   // [4,512,256]
  const float* W_a = (const float*)d_in[1];   // [256,256]
  const float* V_a = (const float*)d_in[2];   // [256,256]
  const float* w_a = (const float*)d_in[3];   // [256]
  float* out = (float*)d_out;                 // [4,512,256]

  float* keyv = (float*)d_ws;                       // 2048*256 f32 = 2 MiB
  float* qryv = keyv + (size_t)BATCH * TSEQ * KDIM; // next 2 MiB

  proj_kernel<<<512, 256, 0, stream>>>(h_e, W_a, V_a, keyv, qryv);
  attn_kernel<<<BATCH * (TSEQ / 16), 256, 0, stream>>>(h_e, keyv, qryv, w_a, out);
}